// CycleMixDisentangledCNNGRU_44461501448604
// MI455X (gfx1250) — compile-verified
//
#include <hip/hip_runtime.h>
#include <stdint.h>

#define B_ROWS      8192
#define DIM         4096
#define NCH         8
#define F4_PER_ROW  (DIM / 4)          // 1024 float4 slots per row
#define F4_PER_GRP  (F4_PER_ROW / NCH) // 128 float4 slots per channel group
#define ALPHA       0.3f

typedef float vf4 __attribute__((ext_vector_type(4)));  // native 16B vector

// ---------- deterministic counter-based RNG (splitmix64 finalizer) ----------
__device__ __forceinline__ uint64_t mix64(uint64_t x) {
    x += 0x9E3779B97F4A7C15ULL;
    x = (x ^ (x >> 30)) * 0xBF58476D1CE4E5B9ULL;
    x = (x ^ (x >> 27)) * 0x94D049BB133111EBULL;
    return x ^ (x >> 31);
}
__device__ __forceinline__ float u01(uint64_t r) {
    // uniform in (0, 1]
    return (float)((uint32_t)(r >> 40) + 1u) * (1.0f / 16777216.0f);
}

// ---------------------------------------------------------------------------
// Kernel 1: per-(channel,row) lambda ~ Beta(a,a) and donor index uniform over
// rows with a different subject label (== Gumbel-argmax over masked scores).
// ---------------------------------------------------------------------------
__global__ void cm_setup(const int* __restrict__ labels,
                         int*   __restrict__ perm,   // [NCH * B_ROWS]
                         float* __restrict__ lam) {  // [NCH * B_ROWS]
    int idx = blockIdx.x * blockDim.x + threadIdx.x;
    if (idx >= NCH * B_ROWS) return;
    const int i  = idx & (B_ROWS - 1);
    const int my = labels[i];
    const uint64_t ctr = 0xC0FFEE1234ULL + (uint64_t)idx * 0xD1B54A32D192ED03ULL;

    // Johnk's algorithm: X=U^(1/a), Y=V^(1/a); accept if X+Y<=1 -> lam=X/(X+Y)
    float l = 0.5f;
    #pragma unroll 1
    for (int t = 0; t < 48; ++t) {
        float u = u01(mix64(ctr + (uint64_t)(2 * t)));
        float v = u01(mix64(ctr + (uint64_t)(2 * t + 1)));
        float x = __powf(u, 1.0f / ALPHA);
        float y = __powf(v, 1.0f / ALPHA);
        float s = x + y;
        if (s <= 1.0f) {             // Beta(0.3,0.3) is heavily bimodal: if both
            l = (s > 0.0f) ? (x / s) // underflow, the mass sits at {0,1}
                           : ((u > v) ? 1.0f : 0.0f);
            break;
        }
    }

    // Rejection-sample a donor with a different label; fallback to self
    // (matches the reference's has_cand fallback).
    int j = i;
    #pragma unroll 1
    for (int t = 0; t < 64; ++t) {
        uint64_t r = mix64(ctr + 4096u + (uint64_t)t);
        int cand = (int)(r & (uint64_t)(B_ROWS - 1));
        if (labels[cand] != my) { j = cand; break; }
    }
    perm[idx] = j;
    lam[idx]  = l;
}

// ---------------------------------------------------------------------------
// Kernel 2: one workgroup (256 threads = 8 wave32) per row.
// Stage the composite donor row (8 slices from 8 donor rows, 16 KB) into LDS
// via the CDNA5 async data-mover (global_load_async_to_lds_b128, ASYNCcnt),
// overlap the wave's own b128 self loads with the async wait, then blend
// out = lam*self + (1-lam)*donor with non-temporal b128 stores (keep the
// 192 MB L2 resident with z so donor gathers stay L2-hit).
// ---------------------------------------------------------------------------
__global__ void __launch_bounds__(256)
cm_mix(const float* __restrict__ z,
       const int*   __restrict__ perm,
       const float* __restrict__ lam,
       float*       __restrict__ out) {
    __shared__ __align__(16) float donor[DIM];   // 16 KB staging buffer
    __shared__ float s_lam[NCH];
    __shared__ int   s_perm[NCH];

    const int row = blockIdx.x;
    const int tid = threadIdx.x;

    if (tid < NCH) {
        s_perm[tid] = perm[tid * B_ROWS + row];
        s_lam[tid]  = lam[tid * B_ROWS + row];
    }
    __syncthreads();

    // LDS byte offset of the staging buffer (flat LDS addr low 32 bits = offset)
    const uint32_t lds_base = (uint32_t)(uintptr_t)(&donor[0]);

    // Issue 4 async 16B copies per lane: slot f4 of the composite donor row
    // comes from z[perm[k]] where k = f4 / 128. Per-lane global addresses,
    // per-lane LDS destinations — GV-mode async-to-LDS.
    #pragma unroll
    for (int s = 0; s < 4; ++s) {
        const int f4 = tid + s * 256;
        const int k  = f4 >> 7;                       // 128 float4 per group
        const uint64_t gaddr =
            (uint64_t)(uintptr_t)(z + (size_t)s_perm[k] * DIM + (size_t)f4 * 4);
        const uint32_t ldsa = lds_base + (uint32_t)(f4 * 16);
        asm volatile("global_load_async_to_lds_b128 %0, %1, off"
                     :: "v"(ldsa), "v"(gaddr)
                     : "memory");
    }

    // Overlap: get the wave's own self-row loads in flight while the async
    // data-mover finishes staging the donor slices.
    const vf4* self_row = (const vf4*)(z + (size_t)row * DIM);
    vf4 a[4];
    #pragma unroll
    for (int s = 0; s < 4; ++s) a[s] = self_row[tid + s * 256];

    // Wait for this wave's async transfers, then make them visible block-wide.
    asm volatile("s_wait_asynccnt 0" ::: "memory");
    __syncthreads();

    vf4*       out_row = (vf4*)(out + (size_t)row * DIM);
    const vf4* donor4  = (const vf4*)donor;

    #pragma unroll
    for (int s = 0; s < 4; ++s) {
        const int f4 = tid + s * 256;
        const int k  = f4 >> 7;
        const float l = s_lam[k];
        const vf4   d = donor4[f4];
        vf4 o;
        o.x = fmaf(l, a[s].x - d.x, d.x);   // == l*a + (1-l)*d
        o.y = fmaf(l, a[s].y - d.y, d.y);
        o.z = fmaf(l, a[s].z - d.z, d.z);
        o.w = fmaf(l, a[s].w - d.w, d.w);
        // Write-once stream: non-temporal so L2 stays dedicated to z.
        __builtin_nontemporal_store(o, &out_row[f4]);
    }
}

// ---------------------------------------------------------------------------
extern "C" void kernel_launch(void* const* d_in, const int* in_sizes, int n_in,
                              void* d_out, int out_size, void* d_ws, size_t ws_size,
                              hipStream_t stream) {
    const float* z_style = (const float*)d_in[0];
    const int*   labels  = (const int*)d_in[1];
    float*       out     = (float*)d_out;

    // Workspace layout: perm[NCH*B] (int) then lam[NCH*B] (float) = 512 KB
    int*   perm = (int*)d_ws;
    float* lam  = (float*)(perm + NCH * B_ROWS);

    const int total = NCH * B_ROWS;                 // 65536
    cm_setup<<<(total + 255) / 256, 256, 0, stream>>>(labels, perm, lam);
    cm_mix<<<B_ROWS, 256, 0, stream>>>(z_style, perm, lam, out);
}